// RecurrentTokenizer_7078106104449
// MI455X (gfx1250) — compile-verified
//
#include <hip/hip_runtime.h>
#include <hip/hip_bf16.h>

typedef __attribute__((ext_vector_type(2))) float v2f;
typedef __attribute__((ext_vector_type(8))) float v8f;

#define NB 8
#define HWP 16384
#define CC 256
#define LL 16
#define DD 256
#define NCHUNK 8

__device__ __forceinline__ v8f wmma_f32(v2f a, v2f b, v8f c) {
  // D = A(16x4) * B(4x16) + C, all fp32, wave32
  return __builtin_amdgcn_wmma_f32_16x16x4_f32(false, a, false, b, (short)0, c,
                                               false, false);
}

// ---------------- a = t @ W1.T + b1   (per n: 16x256, K=256) ----------------
__global__ void k_compute_a(const float* __restrict__ t,
                            const float* __restrict__ W1,
                            const float* __restrict__ b1,
                            float* __restrict__ a_out) {
  int n = blockIdx.y;
  int d0 = blockIdx.x * 16;
  int lane = threadIdx.x;
  int hl = lane & 15;
  int kh = (lane >> 4) * 2;
  const float* tn = t + (size_t)n * LL * DD;
  v8f acc = {};
  for (int k = 0; k < DD; k += 4) {
    v2f af, bf;
    af.x = tn[hl * DD + k + kh];
    af.y = tn[hl * DD + k + kh + 1];
    bf.x = W1[(d0 + hl) * DD + k + kh];      // B[k][d] = W1[d][k]
    bf.y = W1[(d0 + hl) * DD + k + kh + 1];
    acc = wmma_f32(af, bf, acc);
  }
  int mh = (lane >> 4) * 8;
  float bias = b1[d0 + hl];
  float* ao = a_out + (size_t)n * LL * DD;
#pragma unroll
  for (int r = 0; r < 8; ++r)
    ao[(r + mh) * DD + d0 + hl] = acc[r] + bias;
}

// ---------------- v = a @ W2   (per n: 16x256, K=256) ----------------
__global__ void k_compute_v(const float* __restrict__ a_in,
                            const float* __restrict__ W2,
                            float* __restrict__ v_out) {
  int n = blockIdx.y;
  int c0 = blockIdx.x * 16;
  int lane = threadIdx.x;
  int hl = lane & 15;
  int kh = (lane >> 4) * 2;
  const float* an = a_in + (size_t)n * LL * DD;
  v8f acc = {};
  for (int k = 0; k < DD; k += 4) {
    v2f af, bf;
    af.x = an[hl * DD + k + kh];
    af.y = an[hl * DD + k + kh + 1];
    bf.x = W2[(k + kh) * CC + c0 + hl];      // B[k][c] = W2[k][c]
    bf.y = W2[(k + kh + 1) * CC + c0 + hl];
    acc = wmma_f32(af, bf, acc);
  }
  int mh = (lane >> 4) * 8;
  float* vo = v_out + (size_t)n * LL * CC;
#pragma unroll
  for (int r = 0; r < 8; ++r)
    vo[(r + mh) * CC + c0 + hl] = acc[r];
}

// ---------------- s0[n][l] = a[n][l] . b2 ----------------
__global__ void k_s0(const float* __restrict__ a_in,
                     const float* __restrict__ b2, float* __restrict__ s0) {
  int idx = blockIdx.x * blockDim.x + threadIdx.x;
  if (idx >= NB * LL) return;
  const float* ar = a_in + (size_t)idx * DD;
  float s = 0.f;
  for (int d = 0; d < DD; ++d) s += ar[d] * b2[d];
  s0[idx] = s;
}

// -------- scoresT[n][l][p] = (x[n] @ v[n].T + s0), stored L-major --------
__global__ void k_scores(const float* __restrict__ x,
                         const float* __restrict__ v_in,
                         const float* __restrict__ s0,
                         float* __restrict__ scoresT) {
  int n = blockIdx.y;
  int wave = threadIdx.x >> 5;
  int lane = threadIdx.x & 31;
  int p0 = (blockIdx.x * 8 + wave) * 16;
  int hl = lane & 15;
  int kh = (lane >> 4) * 2;
  const float* xn = x + (size_t)n * HWP * CC;
  const float* vn = v_in + (size_t)n * LL * CC;
  v8f acc = {};
  for (int k = 0; k < CC; k += 4) {
    v2f af, bf;
    af.x = xn[(size_t)(p0 + hl) * CC + k + kh];
    af.y = xn[(size_t)(p0 + hl) * CC + k + kh + 1];
    bf.x = vn[hl * CC + k + kh];             // B[k][l] = v[l][k]
    bf.y = vn[hl * CC + k + kh + 1];
    acc = wmma_f32(af, bf, acc);
  }
  int mh = (lane >> 4) * 8;                  // M = p index within tile
  float s0v = s0[n * LL + hl];               // N = l = hl
  float* st = scoresT + (size_t)n * LL * HWP;
#pragma unroll
  for (int r = 0; r < 8; ++r)
    st[(size_t)hl * HWP + p0 + r + mh] = acc[r] + s0v;
}

// -------- softmax stats over p: rowmax, 1/sum(exp) per (n,l) --------
__global__ void k_stats(const float* __restrict__ scoresT,
                        float* __restrict__ mrow, float* __restrict__ inv_s) {
  int row = blockIdx.x;                      // n*L + l
  const float* sr = scoresT + (size_t)row * HWP;
  __shared__ float red[256];
  float mx = -3.4e38f;
  for (int p = threadIdx.x; p < HWP; p += 256) mx = fmaxf(mx, sr[p]);
  red[threadIdx.x] = mx;
  __syncthreads();
  for (int off = 128; off; off >>= 1) {
    if (threadIdx.x < off)
      red[threadIdx.x] = fmaxf(red[threadIdx.x], red[threadIdx.x + off]);
    __syncthreads();
  }
  mx = red[0];
  __syncthreads();
  float sum = 0.f;
  for (int p = threadIdx.x; p < HWP; p += 256) sum += __expf(sr[p] - mx);
  red[threadIdx.x] = sum;
  __syncthreads();
  for (int off = 128; off; off >>= 1) {
    if (threadIdx.x < off) red[threadIdx.x] += red[threadIdx.x + off];
    __syncthreads();
  }
  if (threadIdx.x == 0) {
    mrow[row] = mx;
    inv_s[row] = 1.0f / red[0];
  }
}

// ---- u_part[n][ch] = exp(scoresT chunk - m) @ x chunk   (16x256, K=HW/8) ----
__global__ void k_u(const float* __restrict__ x,
                    const float* __restrict__ scoresT,
                    const float* __restrict__ mrow,
                    float* __restrict__ u_part) {
  int n = blockIdx.y;
  int ct = blockIdx.x & 15;                  // c tile
  int ch = blockIdx.x >> 4;                  // p chunk
  int c0 = ct * 16;
  int lane = threadIdx.x;
  int hl = lane & 15;
  int kh = (lane >> 4) * 2;
  const float* xn = x + (size_t)n * HWP * CC;
  const float* st = scoresT + (size_t)n * LL * HWP + (size_t)hl * HWP;
  float mv = mrow[n * LL + hl];              // A row = l = hl
  v8f acc = {};
  int pbeg = ch * (HWP / NCHUNK), pend = pbeg + HWP / NCHUNK;
  for (int p = pbeg; p < pend; p += 4) {
    v2f af, bf;
    af.x = __expf(st[p + kh] - mv);          // unnormalized attn on the fly
    af.y = __expf(st[p + kh + 1] - mv);
    bf.x = xn[(size_t)(p + kh) * CC + c0 + hl];
    bf.y = xn[(size_t)(p + kh + 1) * CC + c0 + hl];
    acc = wmma_f32(af, bf, acc);
  }
  int mh = (lane >> 4) * 8;                  // M = l
  float* up = u_part + ((size_t)n * NCHUNK + ch) * LL * CC;
#pragma unroll
  for (int r = 0; r < 8; ++r)
    up[(size_t)(r + mh) * CC + c0 + hl] = acc[r];
}

// ---- u[n][l][c] = inv_s[n,l] * sum_ch u_part ----
__global__ void k_reduce_u(const float* __restrict__ u_part,
                           const float* __restrict__ inv_s,
                           float* __restrict__ u_out) {
  int idx = blockIdx.x * blockDim.x + threadIdx.x;
  if (idx >= NB * LL * CC) return;
  int n = idx / (LL * CC);
  int lc = idx % (LL * CC);
  int l = lc / CC;
  float s = 0.f;
#pragma unroll
  for (int ch = 0; ch < NCHUNK; ++ch)
    s += u_part[((size_t)n * NCHUNK + ch) * LL * CC + lc];
  u_out[idx] = s * inv_s[n * LL + l];
}

// ---------------- t_new = u @ W2.T + b2 ----------------
__global__ void k_out(const float* __restrict__ u_in,
                      const float* __restrict__ W2,
                      const float* __restrict__ b2, float* __restrict__ out) {
  int n = blockIdx.y;
  int d0 = blockIdx.x * 16;
  int lane = threadIdx.x;
  int hl = lane & 15;
  int kh = (lane >> 4) * 2;
  const float* un = u_in + (size_t)n * LL * CC;
  v8f acc = {};
  for (int k = 0; k < CC; k += 4) {
    v2f af, bf;
    af.x = un[hl * CC + k + kh];
    af.y = un[hl * CC + k + kh + 1];
    bf.x = W2[(d0 + hl) * CC + k + kh];      // B[k][d] = W2[d][k]
    bf.y = W2[(d0 + hl) * CC + k + kh + 1];
    acc = wmma_f32(af, bf, acc);
  }
  int mh = (lane >> 4) * 8;
  float bias = b2[d0 + hl];
  float* on = out + (size_t)n * LL * DD;
#pragma unroll
  for (int r = 0; r < 8; ++r)
    on[(size_t)(r + mh) * DD + d0 + hl] = acc[r] + bias;
}

extern "C" void kernel_launch(void* const* d_in, const int* in_sizes, int n_in,
                              void* d_out, int out_size, void* d_ws,
                              size_t ws_size, hipStream_t stream) {
  const float* x = (const float*)d_in[0];
  const float* t = (const float*)d_in[1];
  const float* W1 = (const float*)d_in[2];
  const float* b1 = (const float*)d_in[3];
  const float* W2 = (const float*)d_in[4];
  const float* b2 = (const float*)d_in[5];
  float* out = (float*)d_out;

  float* ws = (float*)d_ws;
  float* a = ws;                                      // N*16*256
  float* v = a + NB * LL * DD;                        // N*16*256
  float* s0 = v + NB * LL * CC;                       // N*16
  float* mrow = s0 + NB * LL;                         // N*16
  float* inv_s = mrow + NB * LL;                      // N*16
  float* scoresT = inv_s + NB * LL;                   // N*16*16384
  float* u_part = scoresT + (size_t)NB * LL * HWP;    // N*8*16*256
  float* u = u_part + (size_t)NB * NCHUNK * LL * CC;  // N*16*256

  k_compute_a<<<dim3(DD / 16, NB), 32, 0, stream>>>(t, W1, b1, a);
  k_compute_v<<<dim3(CC / 16, NB), 32, 0, stream>>>(a, W2, v);
  k_s0<<<1, 128, 0, stream>>>(a, b2, s0);
  k_scores<<<dim3(HWP / 16 / 8, NB), 256, 0, stream>>>(x, v, s0, scoresT);
  k_stats<<<NB * LL, 256, 0, stream>>>(scoresT, mrow, inv_s);
  k_u<<<dim3(16 * NCHUNK, NB), 32, 0, stream>>>(x, scoresT, mrow, u_part);
  k_reduce_u<<<(NB * LL * CC + 255) / 256, 256, 0, stream>>>(u_part, inv_s, u);
  k_out<<<dim3(DD / 16, NB), 32, 0, stream>>>(u, W2, b2, out);
}